// Encoder_34746285425414
// MI455X (gfx1250) — compile-verified
//
#include <hip/hip_runtime.h>
#include <hip/hip_bf16.h>

typedef __attribute__((ext_vector_type(2))) float v2f;
typedef __attribute__((ext_vector_type(8))) float v8f;

#define TM 128        // rows per block tile (4 waves x 32 rows)
#define TN 64         // cols per block tile
#define KC 64         // K chunk staged in LDS
#define KC2 (KC / 2)  // k-pairs per chunk
#define ASTR (KC2 + 1)   // As2 row stride in float2 (33)
#define WSTR (TN + 16)   // Ws2 row stride in float2 (80): 2*80 % 64 == 32 -> halves disjoint

// ---------------------------------------------------------------------------
// GEMM:  Y[M x Nc] = (A1 (+A2)) [M x K] @ W [K x Nc] + bias, with fused
// per-column sum / sumsq accumulation for the following batchnorm.
// f32 WMMA 16x16x4. 128 threads = 4 waves; each wave owns a 32x64 output
// tile (2 row strips x 4 col tiles, 8 accumulators), so every B fragment
// (single ds_load_b64 from the k-pair packed tile) feeds two WMMAs.
// ---------------------------------------------------------------------------
template <bool ADD2>
__global__ __launch_bounds__(128) void gemm_stats_kernel(
    const float* __restrict__ A1, const float* __restrict__ A2,
    const float* __restrict__ W, const float* __restrict__ bias,
    float* __restrict__ Y, float* __restrict__ stats,
    int M, int K, int Nc)
{
    __shared__ v2f As2[TM][ASTR];   // As2[m][kk] = (A[m][2kk], A[m][2kk+1])
    __shared__ v2f Ws2[KC2][WSTR];  // Ws2[kk][n] = (W[2kk][n], W[2kk+1][n])

    const int tid  = threadIdx.x;
    const int lane = tid & 31;
    const int wave = tid >> 5;
    const int rowbase = blockIdx.x * TM;
    const int colbase = blockIdx.y * TN;
    const int ln15 = lane & 15;
    const int hi   = lane >> 4;          // 0 for lanes 0-15, 1 for 16-31

    // accumulators: 2 row strips x 4 col tiles, init with bias
    v8f acc0[4], acc1[4];
#pragma unroll
    for (int ct = 0; ct < 4; ++ct) {
        float b = bias[colbase + ct * 16 + ln15];
#pragma unroll
        for (int j = 0; j < 8; ++j) { acc0[ct][j] = b; acc1[ct][j] = b; }
    }

    for (int kc = 0; kc < K; kc += KC) {
        // stage A chunk as k-pairs (coalesced 8B reads/writes), zero-fill m >= M
        for (int idx = tid; idx < TM * KC2; idx += 128) {
            int m  = idx >> 5;          // / KC2
            int kk = idx & (KC2 - 1);
            int grow = rowbase + m;
            v2f v = {0.f, 0.f};
            if (grow < M) {
                const float2 va = *(const float2*)(A1 + (size_t)grow * K + kc + 2 * kk);
                v.x = va.x; v.y = va.y;
                if (ADD2) {
                    const float2 vb = *(const float2*)(A2 + (size_t)grow * K + kc + 2 * kk);
                    v.x += vb.x; v.y += vb.y;
                }
            }
            As2[m][kk] = v;
        }
        // stage W chunk as k-pairs: two coalesced row reads -> one float2
        for (int idx = tid; idx < KC2 * TN; idx += 128) {
            int kk = idx >> 6;          // / TN
            int n  = idx & (TN - 1);
            v2f v;
            v.x = W[(size_t)(kc + 2 * kk) * Nc + colbase + n];
            v.y = W[(size_t)(kc + 2 * kk + 1) * Nc + colbase + n];
            Ws2[kk][n] = v;
        }
        __syncthreads();

        const int m0 = wave * 32 + ln15;
#pragma unroll 4
        for (int k = 0; k < KC; k += 4) {
            const int kk = (k >> 1) + hi;     // half-wave split: K={0,1} | {2,3}
            v2f a0 = As2[m0][kk];
            v2f a1 = As2[m0 + 16][kk];
#pragma unroll
            for (int ct = 0; ct < 4; ++ct) {
                v2f b = Ws2[kk][ct * 16 + ln15];
                acc0[ct] = __builtin_amdgcn_wmma_f32_16x16x4_f32(
                    false, a0, false, b, (short)0, acc0[ct], false, false);
                acc1[ct] = __builtin_amdgcn_wmma_f32_16x16x4_f32(
                    false, a1, false, b, (short)0, acc1[ct], false, false);
            }
        }
        __syncthreads();
    }

    // epilogue: store + fused per-column sum / sumsq (rows >= M masked out)
    const int mbase = rowbase + wave * 32 + (hi << 3);
#pragma unroll
    for (int ct = 0; ct < 4; ++ct) {
        int col = colbase + ct * 16 + ln15;
        float s = 0.f, s2 = 0.f;
#pragma unroll
        for (int j = 0; j < 8; ++j) {
            int row0 = mbase + j;        // strip 0
            if (row0 < M) {
                float v = acc0[ct][j];
                Y[(size_t)row0 * Nc + col] = v;
                s += v; s2 += v * v;
            }
            int row1 = mbase + 16 + j;   // strip 1
            if (row1 < M) {
                float v = acc1[ct][j];
                Y[(size_t)row1 * Nc + col] = v;
                s += v; s2 += v * v;
            }
        }
        unsafeAtomicAdd(&stats[col], s);
        unsafeAtomicAdd(&stats[Nc + col], s2);
    }
}

// ---------------------------------------------------------------------------
// BatchNorm apply (in place) with optional relu.  stats: [sum | sumsq].
// ---------------------------------------------------------------------------
__global__ __launch_bounds__(256) void bn_apply_kernel(
    float* __restrict__ Y, const float* __restrict__ stats,
    const float* __restrict__ g, const float* __restrict__ be,
    int M, int Nc, int do_relu)
{
    const size_t n4 = ((size_t)M * Nc) >> 2;
    const float invM = 1.f / (float)M;
    for (size_t i = (size_t)blockIdx.x * blockDim.x + threadIdx.x; i < n4;
         i += (size_t)gridDim.x * blockDim.x) {
        int c0 = (int)((i << 2) % (size_t)Nc);
        float4 y = ((float4*)Y)[i];
        float* yp = (float*)&y;
#pragma unroll
        for (int j = 0; j < 4; ++j) {
            int col = c0 + j;
            float m   = stats[col] * invM;
            float var = stats[Nc + col] * invM - m * m;
            float rs  = rsqrtf(var + 1e-5f);
            float v = (yp[j] - m) * rs * g[col] + be[col];
            if (do_relu) v = fmaxf(v, 0.f);
            yp[j] = v;
        }
        ((float4*)Y)[i] = y;
    }
}

// ---------------------------------------------------------------------------
// Node embedding gather: h[n][:] = atom_emb[x[n]][:]
// ---------------------------------------------------------------------------
__global__ __launch_bounds__(256) void embed_kernel(
    const int* __restrict__ x, const float* __restrict__ emb,
    float* __restrict__ h, int N)
{
    int idx = blockIdx.x * blockDim.x + threadIdx.x;   // over N * 32 float4
    if (idx >= N * 32) return;
    int n = idx >> 5;
    int c = idx & 31;
    ((float4*)h)[idx] = ((const float4*)emb)[x[n] * 32 + c];
}

// ---------------------------------------------------------------------------
// Edge message + scatter: one wave32 per edge, float4 per lane (32*4 = 128).
// msg = relu(h[src] + bond_emb[attr]) * w  -> atomic add into agg[dst].
// ---------------------------------------------------------------------------
__global__ __launch_bounds__(256) void edge_msg_kernel(
    const int* __restrict__ ei, const int* __restrict__ eattr,
    const float* __restrict__ ew, const float* __restrict__ h,
    const float* __restrict__ bond, float* __restrict__ agg, int E)
{
    int wid  = (blockIdx.x << 3) | (threadIdx.x >> 5);
    int lane = threadIdx.x & 31;
    if (wid >= E) return;
    int src = ei[wid];
    int dst = ei[E + wid];
    int a   = eattr[wid];
    float w = ew[wid];
    float4 hv = ((const float4*)(h + (size_t)src * 128))[lane];
    float4 bv = ((const float4*)(bond + (size_t)a * 128))[lane];
    float4 m;
    m.x = fmaxf(hv.x + bv.x, 0.f) * w;
    m.y = fmaxf(hv.y + bv.y, 0.f) * w;
    m.z = fmaxf(hv.z + bv.z, 0.f) * w;
    m.w = fmaxf(hv.w + bv.w, 0.f) * w;
    float* p = agg + (size_t)dst * 128 + lane * 4;
    unsafeAtomicAdd(p + 0, m.x);
    unsafeAtomicAdd(p + 1, m.y);
    unsafeAtomicAdd(p + 2, m.z);
    unsafeAtomicAdd(p + 3, m.w);
}

// ---------------------------------------------------------------------------
// Graph pooling: copy h -> out_h and atomic-add rows into out_pool[batch[n]].
// ---------------------------------------------------------------------------
__global__ __launch_bounds__(256) void pool_copy_kernel(
    const float* __restrict__ h, const int* __restrict__ batch,
    float* __restrict__ out_pool, float* __restrict__ out_h, int N)
{
    int wid  = (blockIdx.x << 3) | (threadIdx.x >> 5);
    int lane = threadIdx.x & 31;
    if (wid >= N) return;
    float4 v = ((const float4*)(h + (size_t)wid * 128))[lane];
    ((float4*)(out_h + (size_t)wid * 128))[lane] = v;
    int g = batch[wid];
    float* p = out_pool + (size_t)g * 128 + lane * 4;
    unsafeAtomicAdd(p + 0, v.x);
    unsafeAtomicAdd(p + 1, v.y);
    unsafeAtomicAdd(p + 2, v.z);
    unsafeAtomicAdd(p + 3, v.w);
}

// ---------------------------------------------------------------------------
// Row L2-normalize pooled output. One block of 128 threads per graph row.
// ---------------------------------------------------------------------------
__global__ __launch_bounds__(128) void normalize_kernel(float* pool, int D)
{
    __shared__ float red[128];
    int g = blockIdx.x;
    int t = threadIdx.x;
    float v = pool[(size_t)g * D + t];
    red[t] = v * v;
    __syncthreads();
    for (int s = 64; s > 0; s >>= 1) {
        if (t < s) red[t] += red[t + s];
        __syncthreads();
    }
    float scale = 1.f / fmaxf(sqrtf(red[0]), 1e-12f);
    pool[(size_t)g * D + t] = v * scale;
}

// ---------------------------------------------------------------------------

extern "C" void kernel_launch(void* const* d_in, const int* in_sizes, int n_in,
                              void* d_out, int out_size, void* d_ws, size_t ws_size,
                              hipStream_t stream)
{
    const int N = 100000, E = 1600000, D = 128, L = 5, G = 256;

    const int*   batch    = (const int*)d_in[0];
    const int*   x        = (const int*)d_in[1];
    const int*   ei       = (const int*)d_in[2];   // [2, E]
    const int*   eattr    = (const int*)d_in[3];
    const float* ew       = (const float*)d_in[4];
    const float* atom_emb = (const float*)d_in[5];
    const float* bond_emb = (const float*)d_in[6];
    const float* W1  = (const float*)d_in[7];      // [L, D, 2D]
    const float* b1  = (const float*)d_in[8];
    const float* g1  = (const float*)d_in[9];
    const float* be1 = (const float*)d_in[10];
    const float* W2  = (const float*)d_in[11];     // [L, 2D, D]
    const float* b2  = (const float*)d_in[12];
    const float* g2  = (const float*)d_in[13];
    const float* be2 = (const float*)d_in[14];

    // workspace layout (f32): h[N*D] | agg[N*D] | z1[N*2D] | stats[1024]
    float* h     = (float*)d_ws;
    float* agg   = h + (size_t)N * D;
    float* z1    = agg + (size_t)N * D;
    float* stats = z1 + (size_t)N * 2 * D;

    embed_kernel<<<(N * 32 + 255) / 256, 256, 0, stream>>>(x, atom_emb, h, N);

    const dim3 gemm1_grid((N + TM - 1) / TM, (2 * D) / TN);
    const dim3 gemm2_grid((N + TM - 1) / TM, D / TN);

    for (int i = 0; i < L; ++i) {
        hipMemsetAsync(agg, 0, (size_t)N * D * sizeof(float), stream);
        hipMemsetAsync(stats, 0, 1024 * sizeof(float), stream);

        edge_msg_kernel<<<(E + 7) / 8, 256, 0, stream>>>(
            ei, eattr, ew, h, bond_emb, agg, E);

        gemm_stats_kernel<true><<<gemm1_grid, 128, 0, stream>>>(
            h, agg, W1 + (size_t)i * D * 2 * D, b1 + (size_t)i * 2 * D,
            z1, stats, N, D, 2 * D);

        bn_apply_kernel<<<2048, 256, 0, stream>>>(
            z1, stats, g1 + (size_t)i * 2 * D, be1 + (size_t)i * 2 * D,
            N, 2 * D, 1);

        gemm_stats_kernel<false><<<gemm2_grid, 128, 0, stream>>>(
            z1, nullptr, W2 + (size_t)i * 2 * D * D, b2 + (size_t)i * D,
            h, stats + 512, N, 2 * D, D);

        bn_apply_kernel<<<2048, 256, 0, stream>>>(
            h, stats + 512, g2 + (size_t)i * D, be2 + (size_t)i * D,
            N, D, (i < L - 1) ? 1 : 0);
    }

    float* out_pool = (float*)d_out;              // [G, D]
    float* out_h    = out_pool + (size_t)G * D;   // [N, D]
    hipMemsetAsync(out_pool, 0, (size_t)G * D * sizeof(float), stream);
    pool_copy_kernel<<<(N + 7) / 8, 256, 0, stream>>>(h, batch, out_pool, out_h, N);
    normalize_kernel<<<G, 128, 0, stream>>>(out_pool, D);
}